// CostVolume_63831803953160
// MI455X (gfx1250) — compile-verified
//
#include <hip/hip_runtime.h>

// Cost-volume construction: pure data movement (0 FLOPs, ~377MB streamed out).
// Strategy: stage each (c,h) row pair in LDS via CDNA5 async global->LDS B128
// copies, then fan out 96 disparity-shifted rows with non-temporal B128 stores.

typedef float v4f __attribute__((ext_vector_type(4)));
typedef int   v4i __attribute__((ext_vector_type(4)));

#define LDS_SPACE __attribute__((address_space(3)))
#define GLB_SPACE __attribute__((address_space(1)))

#if defined(__gfx1250__) && __has_builtin(__builtin_amdgcn_global_load_async_to_lds_b128)
#define USE_ASYNC_LDS 1
#else
#define USE_ASYNC_LDS 0
#endif

__launch_bounds__(256)
__global__ void cost_volume_kernel(const float* __restrict__ left,
                                   const float* __restrict__ right,
                                   float* __restrict__ out)
{
    constexpr int C  = 32;
    constexpr int H  = 128;
    constexpr int W  = 240;   // 60 float4 per row
    constexpr int D4 = 48;    // max_disparity / 4

    __shared__ float lds_left[W];          // left row
    __shared__ float lds_rpad[D4 + W];     // [0..47] = 0 pad, [48..287] = right row

    const int      t   = threadIdx.x;
    const unsigned bid = blockIdx.x;       // 0 .. C*H-1
    const unsigned c   = bid / H;
    const unsigned h   = bid % H;
    const unsigned rowbase = (c * H + h) * W;

    // Zero the front pad so right-volume shift+mask needs no branch.
    if (t < D4) lds_rpad[t] = 0.0f;

    // Stage both rows: 120 x B128 async transfers (one per lane).
    if (t < 2 * (W / 4)) {
        const bool isr = (t >= (W / 4));
        const int  q   = isr ? (t - (W / 4)) : t;
        const float* src = (isr ? right : left) + rowbase + q * 4;
        float*       dst = isr ? &lds_rpad[D4 + q * 4] : &lds_left[q * 4];
#if USE_ASYNC_LDS
        __builtin_amdgcn_global_load_async_to_lds_b128(
            (GLB_SPACE v4i*)src, (LDS_SPACE v4i*)dst, /*offset=*/0, /*cpol=*/0);
#else
        *(v4f*)dst = *(const v4f*)src;
#endif
    }
#if USE_ASYNC_LDS
#if __has_builtin(__builtin_amdgcn_s_wait_asynccnt)
    __builtin_amdgcn_s_wait_asynccnt(0);
#else
    asm volatile("s_wait_asynccnt 0" ::: "memory");
#endif
#endif
    __syncthreads();

    // 96 output rows (48 left-volume + 48 right-volume), 60 float4 each.
    // All stores are 16B-aligned, streamed (non-temporal: never re-read).
    for (int idx = t; idx < (2 * D4) * (W / 4); idx += 256) {
        const int row = idx / (W / 4);         // 0..95
        const int q   = idx - row * (W / 4);   // 0..59
        const int w0  = q * 4;

        v4f v;
        unsigned cc, d;
        if (row < D4) {                        // left volume: mask w < d
            d  = (unsigned)row;
            cc = c;
            v.x = (w0 + 0 >= (int)d) ? lds_left[w0 + 0] : 0.0f;
            v.y = (w0 + 1 >= (int)d) ? lds_left[w0 + 1] : 0.0f;
            v.z = (w0 + 2 >= (int)d) ? lds_left[w0 + 2] : 0.0f;
            v.w = (w0 + 3 >= (int)d) ? lds_left[w0 + 3] : 0.0f;
        } else {                               // right volume: shifted, pad = mask
            d  = (unsigned)(row - D4);
            cc = c + C;
            const int b = D4 + w0 - (int)d;    // >= 1, <= 284
            v.x = lds_rpad[b + 0];
            v.y = lds_rpad[b + 1];
            v.z = lds_rpad[b + 2];
            v.w = lds_rpad[b + 3];
        }

        const unsigned off = ((cc * (unsigned)D4 + d) * (unsigned)H + h) * (unsigned)W
                           + (unsigned)w0;
        __builtin_nontemporal_store(v, (v4f*)(out + off));
    }
}

extern "C" void kernel_launch(void* const* d_in, const int* in_sizes, int n_in,
                              void* d_out, int out_size, void* d_ws, size_t ws_size,
                              hipStream_t stream)
{
    const float* left  = (const float*)d_in[0];
    const float* right = (const float*)d_in[1];
    float*       out   = (float*)d_out;

    // One block per (c, h) pair: 32 * 128 = 4096 blocks of 256 threads.
    dim3 grid(32 * 128, 1, 1);
    dim3 block(256, 1, 1);
    hipLaunchKernelGGL(cost_volume_kernel, grid, block, 0, stream, left, right, out);
}